// Disp_61125974557155
// MI455X (gfx1250) — compile-verified
//
#include <hip/hip_runtime.h>
#include <stdint.h>

#define DIN  48
#define HIN  64
#define WIN  128
#define DOUT 193
#define HOUT 256
#define WOUT 512

// Output tile per block: 4 rows x 64 cols of output pixels (one batch slice).
#define TILE_H 4
#define TILE_W 64
// Input footprint: 3 input rows x 18 input cols x 48 depths.
#define IN_H   3
#define IN_W   18
#define PLANE  (IN_H * IN_W)          /* 54  */
#define TILE_ELEMS (DIN * PLANE)      /* 2592 floats = 10368 B */

__global__ __launch_bounds__(256)
void Disp_61125974557155_kernel(const float* __restrict__ x,
                                float* __restrict__ out) {
    extern __shared__ float tile[];   // dynamic LDS: 48 x 3 x 18 f32

    const int wt  = blockIdx.x;   // 0..7   (w tiles)
    const int ht  = blockIdx.y;   // 0..63  (h tiles)
    const int b   = blockIdx.z;   // 0..1
    const int tid = threadIdx.x;  // 0..255

    const int w_base = wt * 16 - 1;   // input col of ww==0 (pre-clamp)
    const int h_base = ht - 1;        // input row of hh==0 (pre-clamp)
    const float* xb = x + (size_t)b * (DIN * HIN * WIN);

    // ---- Stage input tile into LDS via CDNA5 async global->LDS copies ----
    // Edge replication (clamp) is baked into the staged tile so the compute
    // loops below are branch-free.
    for (int e = tid; e < TILE_ELEMS; e += 256) {
        int d  = e / PLANE;
        int r  = e - d * PLANE;
        int hh = r / IN_W;
        int ww = r - hh * IN_W;
        int sh = min(max(h_base + hh, 0), HIN - 1);
        int sw = min(max(w_base + ww, 0), WIN - 1);
        const float* gp = xb + ((size_t)d * HIN + sh) * WIN + sw;
        uint32_t lds_addr = (uint32_t)(uintptr_t)(&tile[e]);
        asm volatile("global_load_async_to_lds_b32 %0, %1, off"
                     :: "v"(lds_addr), "v"(gp) : "memory");
    }
    asm volatile("s_wait_asynccnt 0" ::: "memory");
    __syncthreads();

    // ---- Per-pixel H/W bilinear setup (half-pixel centers, clamped) ----
    const int wl = tid & (TILE_W - 1);
    const int hl = tid >> 6;
    const int w  = wt * TILE_W + wl;
    const int h  = ht * TILE_H + hl;

    float cw = 0.25f * (float)w - 0.375f;
    cw = fminf(fmaxf(cw, 0.0f), (float)(WIN - 1));
    int   w0 = (int)cw;
    float fw = cw - (float)w0;

    float chf = 0.25f * (float)h - 0.375f;
    chf = fminf(fmaxf(chf, 0.0f), (float)(HIN - 1));
    int   h0 = (int)chf;
    float fh = chf - (float)h0;

    const int ww0 = w0 - w_base;      // 0..16
    const int hh0 = h0 - h_base;      // 0..1
    const float* t00 = tile + hh0 * IN_W + ww0;

    // ---- HW-interpolated depth column in registers, pre-scaled by log2(e) ----
    // exp(m - v) == exp2(log2e*m - log2e*v), and both lerp and min commute
    // with a positive scale, so scaling z once (48 muls) removes the per-exp
    // multiply (193 muls) that __expf would otherwise emit.
    const float LOG2E = 1.4426950408889634f;
    float z[DIN];
    #pragma unroll
    for (int d = 0; d < DIN; ++d) {
        const float* p = t00 + d * PLANE;
        float a  = p[0];
        float bb = p[1];
        float c  = p[IN_W];
        float ee = p[IN_W + 1];
        float lo = fmaf(fw, bb - a, a);
        float hi = fmaf(fw, ee - c, c);
        z[d] = fmaf(fh, hi - lo, lo) * LOG2E;
    }

    // Stability shift: m = min_d z[d] <= every interpolated value (convex
    // combination), so every exponent (m - v) <= 0. One pass suffices.
    float m = z[0];
    #pragma unroll
    for (int d = 1; d < DIN; ++d) m = fminf(m, z[d]);

    // ---- Fused depth upsample (48 -> 193) + softmax(-v) + E[d] ----
    // c_d = (d+0.5)*48/193 - 0.5 = (96d - 145)/386, clamped to [0,47].
    // Integer math + full unroll => d0/d1/fd are compile-time constants,
    // so z[d0]/z[d1] are static register reads (no scratch).
    float sum = 0.0f, wsum = 0.0f;
    #pragma unroll
    for (int d = 0; d < DOUT; ++d) {
        int   num = 96 * d - 145;
        int   d0  = (num < 0) ? 0 : (num / 386);
        int   rem = (num < 0) ? 0 : (num - d0 * 386);
        int   d1  = (d0 + 1 < DIN) ? (d0 + 1) : (DIN - 1);
        float fd  = (float)rem * (1.0f / 386.0f);
        float v   = fmaf(fd, z[d1] - z[d0], z[d0]);
        float e   = __builtin_amdgcn_exp2f(m - v);   // native v_exp_f32
        sum  += e;
        wsum  = fmaf((float)d, e, wsum);
    }

    out[((size_t)b * HOUT + h) * WOUT + w] = wsum / sum;
}

extern "C" void kernel_launch(void* const* d_in, const int* in_sizes, int n_in,
                              void* d_out, int out_size, void* d_ws, size_t ws_size,
                              hipStream_t stream) {
    const float* x = (const float*)d_in[0];
    float* out = (float*)d_out;
    dim3 grid(WOUT / TILE_W, HOUT / TILE_H, 2);   // 8 x 64 x 2
    dim3 block(256);
    size_t shmem = TILE_ELEMS * sizeof(float);    // 10368 B
    Disp_61125974557155_kernel<<<grid, block, shmem, stream>>>(x, out);
}